// Decoder_20933670601233
// MI455X (gfx1250) — compile-verified
//
#include <hip/hip_runtime.h>

// ---------------------------------------------------------------------------
// SIR RK4 batch integrator for MI455X (gfx1250).
//   - 1 thread = 1 trajectory (B = 65536), wave32, 256 threads/block.
//   - 25 chunks x 8 time points; states staged in LDS, then drained to
//     global memory with coalesced 16B stores.
//   - Drain uses CDNA5 GLOBAL_STORE_ASYNC_FROM_LDS_B128 (ASYNCcnt):
//     double-buffered LDS + s_wait_asynccnt overlaps integration of chunk
//     c+1 with the async drain of chunk c.
// ---------------------------------------------------------------------------

#define BLK     256
#define CH      8           // time points per chunk
#define NCHUNK  25          // 200 / 8
#define STRIDE  28          // floats per trajectory per buffer (24 + pad, 16B aligned)

#if __has_builtin(__builtin_amdgcn_global_store_async_from_lds_b128)
#define ASYNC_STORE 1
#else
#define ASYNC_STORE 0
#endif

// Builtin signature (from clang diagnostic):
//   void __builtin_amdgcn_global_store_async_from_lds_b128(
//       v4i __device__* dst, v4i __shared__* src, imm int offset, imm int cpol)
typedef int v4i __attribute__((vector_size(16)));
typedef __attribute__((address_space(1))) v4i gv4i;   // global (AS1)
typedef __attribute__((address_space(3))) v4i sv4i;   // LDS (AS3)

template <int N>
__device__ __forceinline__ void wait_async() {
#if ASYNC_STORE
#if __has_builtin(__builtin_amdgcn_s_wait_asynccnt)
    __builtin_amdgcn_s_wait_asynccnt(N);
#else
    asm volatile("s_wait_asynccnt %0" :: "i"(N) : "memory");
#endif
#endif
}

__global__ __launch_bounds__(BLK) void sir_rk4_kernel(
    const float4* __restrict__ params, float* __restrict__ out, int B) {

    __shared__ float smem[2][BLK * STRIDE];

    const int tid = threadIdx.x;
    const int b0  = blockIdx.x * BLK;
    const int b   = b0 + tid;

    const float dt = 100.0f / 199.0f;
    const float h2 = 0.5f * dt;
    const float h6 = dt * (1.0f / 6.0f);

    float beta = 0.f, gamma = 0.f, S = 0.f, I = 0.f, R = 0.f;
    if (b < B) {
        float4 p = params[b];          // global_load_b128, fully coalesced
        beta = p.x; gamma = p.y; S = p.z; I = p.w;
        R = 1.0f - S - I;
    }

    for (int c = 0; c < NCHUNK; ++c) {
        const int pbuf = c & 1;
        if (c >= 2) {
            // Buffer `pbuf` was drained at chunk c-2. Async stores complete
            // in order per wave: waiting down to <=6 outstanding forces the
            // chunk c-2 batch (6 stores) done; barrier extends that to all
            // waves before anyone overwrites the buffer.
            wait_async<6>();
            __syncthreads();
        }

        float* buf = &smem[pbuf][tid * STRIDE];
#pragma unroll
        for (int s = 0; s < CH; ++s) {
            if (!(c == 0 && s == 0)) {
                // one RK4 step of  S' = -b*S*I,  I' = b*S*I - g*I,  R' = g*I
                float bSI = beta * S * I;
                float gI  = gamma * I;
                float k1S = -bSI, k1I = bSI - gI, k1R = gI;

                float Sa = S + h2 * k1S, Ia = I + h2 * k1I;
                float bSIa = beta * Sa * Ia, gIa = gamma * Ia;
                float k2S = -bSIa, k2I = bSIa - gIa, k2R = gIa;

                float Sb = S + h2 * k2S, Ib = I + h2 * k2I;
                float bSIb = beta * Sb * Ib, gIb = gamma * Ib;
                float k3S = -bSIb, k3I = bSIb - gIb, k3R = gIb;

                float Sc = S + dt * k3S, Ic = I + dt * k3I;
                float bSIc = beta * Sc * Ic, gIc = gamma * Ic;
                float k4S = -bSIc, k4I = bSIc - gIc, k4R = gIc;

                S += h6 * (k1S + 2.0f * (k2S + k3S) + k4S);
                I += h6 * (k1I + 2.0f * (k2I + k3I) + k4I);
                R += h6 * (k1R + 2.0f * (k2R + k3R) + k4R);
            }
            buf[s * 3 + 0] = S;
            buf[s * 3 + 1] = I;
            buf[s * 3 + 2] = R;
        }
        __syncthreads();   // all 256 trajectories of this chunk visible in LDS

        // Drain: remap (traj q, float4 j) -> lanes so consecutive lanes write
        // contiguous 96B runs. 256 traj * 6 float4 = 1536 stores, 6/thread.
#pragma unroll
        for (int it = 0; it < 6; ++it) {
            const int L = it * BLK + tid;
            const int q = L / 6;
            const int j = L - q * 6;
            if (b0 + q < B) {
                const size_t goff =
                    (size_t)(b0 + q) * 600 + (size_t)c * 24 + (size_t)j * 4;
                float*       gp = out + goff;                       // 16B aligned
                const float* lp = &smem[pbuf][q * STRIDE + j * 4];  // 16B aligned
#if ASYNC_STORE
                __builtin_amdgcn_global_store_async_from_lds_b128(
                    (gv4i*)gp, (sv4i*)lp, 0, 0);
#else
                const float4 v = *reinterpret_cast<const float4*>(lp);
                *reinterpret_cast<float4*>(gp) = v;
#endif
            }
        }
        // no wait here: next chunk integrates into the other buffer while the
        // async engine drains this one. S_ENDPGM's implicit wait-idle covers
        // the final two chunks.
    }
}

extern "C" void kernel_launch(void* const* d_in, const int* in_sizes, int n_in,
                              void* d_out, int out_size, void* d_ws, size_t ws_size,
                              hipStream_t stream) {
    (void)n_in; (void)out_size; (void)d_ws; (void)ws_size;
    const int B = in_sizes[0] / 4;                   // params is (B, 4) fp32
    const float4* params = (const float4*)d_in[0];
    float* out = (float*)d_out;
    const int grid = (B + BLK - 1) / BLK;
    sir_rk4_kernel<<<grid, BLK, 0, stream>>>(params, out, B);
}